// MetaPosterior_5153960755518
// MI455X (gfx1250) — compile-verified
//
#include <hip/hip_runtime.h>

// MetaPosterior for MI455X (gfx1250, wave32).
// One 256-thread (8-wave) block per (perm,k) pair: gather [P_mm|P_mr] into LDS,
// Gauss-Jordan -> X = P_mm^-1 P_mr in place, then X @ diff via
// V_WMMA_F32_16X16X4_F32 tiles (16 tiles x 16 K-slices), fused log-density
// reduction. Deterministic: tree reductions only, no float atomics.

#define Dn      128
#define Mh      64
#define NUMK    16
#define WSTRIDE 130                      // 64x130 f32, pad to dodge bank conflicts
#define LOG2PI  1.8378770664093453f
#define JITTER  0.01f

typedef __attribute__((ext_vector_type(2))) float v2f;
typedef __attribute__((ext_vector_type(8))) float v8f;

__launch_bounds__(256)
__global__ void meta_pair_kernel(const float* __restrict__ meta_theta,
                                 const float* __restrict__ grads,
                                 const float* __restrict__ m_ks,
                                 const int*   __restrict__ perms,
                                 float* __restrict__ partials) {
  __shared__ float W[Mh * WSTRIDE];   // augmented [P_mm | P_mr] -> [I | X]
  __shared__ int   idx[Dn];           // perm row
  __shared__ float tp[Dn];            // meta_theta[perm]
  __shared__ float mp[Dn];            // mk[perm]
  __shared__ float ppred[Mh];         // diag(P_mm) incl. jitter
  __shared__ float colp[Mh];          // pivot-column snapshot for GJ
  __shared__ float red[256];

  const int t = threadIdx.x;
  const int b = blockIdx.x;                 // pair id = perm_row*NUMK + k
  const int k = b & (NUMK - 1);
  const float* __restrict__ gk = grads + (size_t)k * Dn * Dn;
  const float* __restrict__ mk = m_ks + (size_t)k * Dn;
  const int*   __restrict__ pr = perms + (size_t)b * Dn;

  // ---- load perm row + permuted vectors (L2-resident gathers) ----
  if (t < Dn) {
    int pv = pr[t];
    idx[t] = pv;
    tp[t]  = meta_theta[pv];
    mp[t]  = mk[pv];
  }
  __syncthreads();

  // ---- gather augmented matrix: W[i][j] = grad[k][perm[i]][perm[j]] (+jitter on diag) ----
  for (int e = t; e < Mh * Dn; e += 256) {
    int i = e >> 7;          // 0..63  (row of P = grad[im])
    int j = e & 127;         // 0..127 (cols: [im | ir])
    float g = gk[(size_t)idx[i] * Dn + idx[j]];
    if (i == j) g += JITTER;
    W[i * WSTRIDE + j] = g;
  }
  __syncthreads();

  if (t < Mh) ppred[t] = W[t * WSTRIDE + t];
  __syncthreads();

  // ---- Gauss-Jordan (no pivoting: P_mm is SPD + jitter) ----
  for (int p = 0; p < Mh; ++p) {
    float pinv = 1.0f / W[p * WSTRIDE + p];      // broadcast LDS read
    if (t < Mh) colp[t] = W[t * WSTRIDE + p];    // snapshot multipliers
    __syncthreads();
    if (t < Dn) W[p * WSTRIDE + t] *= pinv;      // scale pivot row (128 wide)
    __syncthreads();
    {
      int c = t & 127;                           // column
      float pc = W[p * WSTRIDE + c];             // scaled pivot-row element
      for (int r = (t >> 7); r < Mh; r += 2) {   // 2 threads/column over 64 rows
        if (r != p) W[r * WSTRIDE + c] -= colp[r] * pc;
      }
    }
    __syncthreads();
  }
  // Now W[i][64+l] = X[i][l] = (P_mm^-1 P_mr)[i][l].

  // ---- XD = X @ diff via f32 WMMA; fused log-density ----
  // diff[l][j] = tp[64+l] - mp[64+j]  (synthesized per B-fragment)
  // log_p[i][j] = -.5*log2pi + .5*log(ppred[j]) - .5*ppred[j]*(tp[i] - mp[j] + XD[i][j])^2
  float lsum = 0.0f;
  const int wv   = t >> 5;
  const int lane = t & 31;
  const int half = lane >> 4;   // 0: lanes 0-15, 1: lanes 16-31
  const int l16  = lane & 15;

  for (int tt = wv; tt < 16; tt += 8) {          // uniform per wave -> EXEC all-ones
    const int r0 = (tt >> 2) << 4;               // tile row origin
    const int c0 = (tt & 3) << 4;                // tile col origin
    const int arow = r0 + l16;                   // A fragment row (M) for this lane
    const int bcol = c0 + l16;                   // B/C fragment col (N) for this lane
    const float vb = mp[Mh + bcol];              // mk[ir[bcol]]

    v8f acc = {0.f, 0.f, 0.f, 0.f, 0.f, 0.f, 0.f, 0.f};
    for (int kk = 0; kk < Mh; kk += 4) {
      const int ka = kk + (half ? 2 : 0);        // ISA 16x4 A layout: K 0/1 low, 2/3 high
      v2f a, bf;
      a.x  = W[arow * WSTRIDE + Mh + ka];        // X[arow][ka]
      a.y  = W[arow * WSTRIDE + Mh + ka + 1];    // X[arow][ka+1]
      bf.x = tp[Mh + ka]     - vb;               // diff[ka][bcol]
      bf.y = tp[Mh + ka + 1] - vb;               // diff[ka+1][bcol]
      acc = __builtin_amdgcn_wmma_f32_16x16x4_f32(
          false, a, false, bf, (short)0, acc, false, false);
    }

    const float ppc = ppred[bcol];               // column-broadcast diag(P_mm)
    const float mpc = mp[bcol];                  // column-broadcast mk[im[col]]
    const float lgc = 0.5f * __logf(ppc) - 0.5f * LOG2PI;
#pragma unroll
    for (int r = 0; r < 8; ++r) {
      const int row = r0 + r + (half ? 8 : 0);   // C layout: M split at lane 16
      const float e = tp[row] - mpc + acc[r];    // theta[:M] - m_pred
      lsum += lgc - 0.5f * ppc * e * e;
    }
  }

  // ---- deterministic block reduction ----
  red[t] = lsum;
  __syncthreads();
  for (int s = 128; s > 0; s >>= 1) {
    if (t < s) red[t] += red[t + s];
    __syncthreads();
  }
  if (t == 0) partials[b] = red[0];
}

__launch_bounds__(256)
__global__ void meta_final_kernel(const float* __restrict__ meta_theta,
                                  const float* __restrict__ partials,
                                  int nPart, float* __restrict__ out) {
  __shared__ float sa[256];  // pair-sum partials
  __shared__ float sb[256];  // th @ th
  const int t = threadIdx.x;
  float s = 0.0f;
  for (int i = t; i < nPart; i += 256) s += partials[i];   // fixed-order strided sum
  float d = 0.0f;
  if (t < Dn) { float th = meta_theta[t]; d = th * th; }
  sa[t] = s; sb[t] = d;
  __syncthreads();
  for (int st = 128; st > 0; st >>= 1) {
    if (t < st) { sa[t] += sa[t + st]; sb[t] += sb[t + st]; }
    __syncthreads();
  }
  if (t == 0) {
    // prior_lp = -0.5*(D*log2pi + D*log(alpha) + th@th/alpha), alpha = 1
    const float prior = -0.5f * ((float)Dn * LOG2PI + sb[0]);
    const float loss_pred = sa[0] / 2048.0f;     // MAX_PERM*M*NUM_K = 2*64*16
    const float loss = (1.0f - 1.0f / (float)NUMK) * prior + loss_pred;
    out[0] = -loss;
  }
}

extern "C" void kernel_launch(void* const* d_in, const int* in_sizes, int n_in,
                              void* d_out, int out_size, void* d_ws, size_t ws_size,
                              hipStream_t stream) {
  const float* meta_theta = (const float*)d_in[0];   // (128,1) f32
  const float* grads      = (const float*)d_in[1];   // (16,128,128) f32
  const float* m_ks       = (const float*)d_in[2];   // (16,128) f32
  const int*   perms      = (const int*)d_in[3];     // (254,16,128) i32
  float* out      = (float*)d_out;                   // scalar
  float* partials = (float*)d_ws;                    // nPairs floats of scratch

  const int nPairs = in_sizes[3] / Dn;               // 254*16 = 4064
  meta_pair_kernel<<<nPairs, 256, 0, stream>>>(meta_theta, grads, m_ks, perms, partials);
  meta_final_kernel<<<1, 256, 0, stream>>>(meta_theta, partials, nPairs, out);
}